// GraphAttentionLayer_73607149519395
// MI455X (gfx1250) — compile-verified
//
#include <hip/hip_runtime.h>
#include <math.h>

typedef __attribute__((ext_vector_type(2))) float v2f;
typedef __attribute__((ext_vector_type(4))) float v4f;
typedef __attribute__((ext_vector_type(8))) float v8f;

#define N_NODES 8192
#define KNN 32

// Branchless sorted-ascending top-K insert; static indices only -> arrays
// stay register-resident. Stable on ties (new candidate goes after equal
// values), which together with ascending-index scan order reproduces
// jax.lax.top_k's lower-index-first tie rule.
__device__ __forceinline__ void topk_insert(float cand, int cidx,
                                            float (&bval)[KNN],
                                            int (&bidx)[KNN]) {
  if (cand < bval[KNN - 1]) {
#pragma unroll
    for (int t = KNN - 1; t >= 1; --t) {
      bool stay = (bval[t] <= cand);     // element keeps its slot
      bool shift = (bval[t - 1] > cand); // predecessor shifts into slot t
      float nv = stay ? bval[t] : (shift ? bval[t - 1] : cand);
      int ni = stay ? bidx[t] : (shift ? bidx[t - 1] : cidx);
      bval[t] = nv;
      bidx[t] = ni;
    }
    if (bval[0] > cand) {
      bval[0] = cand;
      bidx[0] = cidx;
    }
  }
}

// ---------------------------------------------------------------------------
// Phase 1: per-node transform. out = clip(feat@W^T,-1,1), sq = |feat|^2,
// s_self = out . a[:7]. featP padded to 8 (col7 = 0); outP = [out 0..6, pol].
// ---------------------------------------------------------------------------
__global__ __launch_bounds__(256) void gat_prep_kernel(
    const float* __restrict__ x, const float* __restrict__ W,
    const float* __restrict__ a, float* __restrict__ featP,
    float* __restrict__ outP, float* __restrict__ sq,
    float* __restrict__ sself) {
  int n = blockIdx.x * blockDim.x + threadIdx.x;
  if (n >= N_NODES) return;
  float f[7];
  float s2 = 0.0f;
#pragma unroll
  for (int j = 0; j < 7; ++j) {
    f[j] = x[n * 8 + j];
    s2 += f[j] * f[j];
  }
  float pol = x[n * 8 + 7];
  sq[n] = s2;
  float ss = 0.0f;
#pragma unroll
  for (int i = 0; i < 7; ++i) {
    float o = 0.0f;
#pragma unroll
    for (int j = 0; j < 7; ++j) o += W[i * 7 + j] * f[j];
    o = fminf(1.0f, fmaxf(-1.0f, o));
    outP[n * 8 + i] = o;
    ss += o * a[i];
  }
  outP[n * 8 + 7] = pol;
#pragma unroll
  for (int j = 0; j < 7; ++j) featP[n * 8 + j] = f[j];
  featP[n * 8 + 7] = 0.0f;
  sself[n] = ss;
}

// ---------------------------------------------------------------------------
// Phase 2: partial KNN. One wave owns 32 rows x one column chunk.
// Gram tiles via V_WMMA_F32_16X16X4_F32 (K padded to 8 -> 2 WMMAs/tile),
// distances staged through LDS (pad 17 -> conflict-free row reads), lane t
// keeps a sorted top-32 for row t in registers. Emits per-chunk sorted lists.
// ---------------------------------------------------------------------------
__global__ __launch_bounds__(128) void gat_knn_kernel(
    const float* __restrict__ featP, const float* __restrict__ sq,
    float* __restrict__ pval, int* __restrict__ pidx, int nchunks,
    int colsPerChunk) {
  __shared__ float stage[4][32][17];
  const int lane = threadIdx.x & 31;
  const int wv = threadIdx.x >> 5;
  const int waveId = blockIdx.x * 4 + wv;
  const int rowblk = waveId / nchunks;
  const int chunk = waveId % nchunks;
  const int rowbase = rowblk * 32;
  const int cbeg = chunk * colsPerChunk;
  const int cend = cbeg + colsPerChunk;
  const int m = lane & 15;     // col index within tile / M index for A
  const int half = lane >> 4;  // 0: K=0,1  1: K=2,3 (WMMA f32 A/B layout)
  const int k0 = half * 2;

  // A operands fixed for this wave -> hoisted out of the column loop.
  v2f a00 = *(const v2f*)(featP + (rowbase + m) * 8 + k0);
  v2f a01 = *(const v2f*)(featP + (rowbase + m) * 8 + k0 + 4);
  v2f a10 = *(const v2f*)(featP + (rowbase + 16 + m) * 8 + k0);
  v2f a11 = *(const v2f*)(featP + (rowbase + 16 + m) * 8 + k0 + 4);

  // sq for the rows this lane's accumulators hold (M = v + 8*half).
  float sqr0[8], sqr1[8];
#pragma unroll
  for (int v = 0; v < 8; ++v) {
    sqr0[v] = sq[rowbase + v + 8 * half];
    sqr1[v] = sq[rowbase + 16 + v + 8 * half];
  }

  float bval[KNN];
  int bidx[KNN];
#pragma unroll
  for (int j = 0; j < KNN; ++j) {
    bval[j] = 3.0e38f;
    bidx[j] = 0;
  }

  for (int cb = cbeg; cb < cend; cb += 16) {
    if (cb + 16 < cend) __builtin_prefetch(featP + (cb + 16 + m) * 8, 0, 0);
    v2f b0 = *(const v2f*)(featP + (cb + m) * 8 + k0);
    v2f b1 = *(const v2f*)(featP + (cb + m) * 8 + k0 + 4);
    float sqc = sq[cb + m];

    v8f d0 = {};
    v8f d1 = {};
    d0 = __builtin_amdgcn_wmma_f32_16x16x4_f32(false, a00, false, b0, (short)0,
                                               d0, false, false);
    d0 = __builtin_amdgcn_wmma_f32_16x16x4_f32(false, a01, false, b1, (short)0,
                                               d0, false, false);
    d1 = __builtin_amdgcn_wmma_f32_16x16x4_f32(false, a10, false, b0, (short)0,
                                               d1, false, false);
    d1 = __builtin_amdgcn_wmma_f32_16x16x4_f32(false, a11, false, b1, (short)0,
                                               d1, false, false);

    // dist2 = sq_r + sq_c - 2g, clamp 0, diag -> +inf; stage row-major.
#pragma unroll
    for (int v = 0; v < 8; ++v) {
      int r0 = v + 8 * half;
      float dd0 = fmaxf(sqr0[v] + sqc - 2.0f * d0[v], 0.0f);
      if (rowbase + r0 == cb + m) dd0 = __builtin_inff();
      stage[wv][r0][m] = dd0;
      int r1 = 16 + v + 8 * half;
      float dd1 = fmaxf(sqr1[v] + sqc - 2.0f * d1[v], 0.0f);
      if (rowbase + r1 == cb + m) dd1 = __builtin_inff();
      stage[wv][r1][m] = dd1;
    }
    __syncthreads();

    // Lane t scans its row's 16 candidates; keep this loop ROLLED so the
    // (branchless) insert body appears once -> small, I-cache-friendly loop.
#pragma unroll 1
    for (int j = 0; j < 16; ++j) {
      float cand = stage[wv][lane][j];
      topk_insert(cand, cb + j, bval, bidx);
    }
    __syncthreads();
  }

  const long long rowOff = (long long)(rowbase + lane) * nchunks + chunk;
#pragma unroll
  for (int j = 0; j < KNN; ++j) {
    pval[rowOff * KNN + j] = bval[j];
    pidx[rowOff * KNN + j] = bidx[j];
  }
}

// ---------------------------------------------------------------------------
// Phase 2b: merge per-chunk sorted lists -> final 32 neighbors per row.
// Chunks scanned in ascending-index order preserves top_k tie ordering.
// ---------------------------------------------------------------------------
__global__ __launch_bounds__(256) void gat_merge_kernel(
    const float* __restrict__ pval, const int* __restrict__ pidx,
    int* __restrict__ nbr, int nchunks) {
  int row = blockIdx.x * blockDim.x + threadIdx.x;
  if (row >= N_NODES) return;
  float bval[KNN];
  int bidx[KNN];
#pragma unroll
  for (int j = 0; j < KNN; ++j) {
    bval[j] = 3.0e38f;
    bidx[j] = 0;
  }
  const long long base = (long long)row * nchunks * KNN;
#pragma unroll 1
  for (int c = 0; c < nchunks; ++c) {
#pragma unroll 1
    for (int j = 0; j < KNN; ++j) {
      float cand = pval[base + c * KNN + j];
      int cidx = pidx[base + c * KNN + j];
      topk_insert(cand, cidx, bval, bidx);
    }
  }
#pragma unroll
  for (int j = 0; j < KNN; ++j) nbr[row * KNN + j] = bidx[j];
}

// ---------------------------------------------------------------------------
// Phase 3: one wave per node, lane k = neighbor k; wave32 shfl softmax +
// aggregate. y = [out(7) | agg(8)] written twice; node N-1 emits att.
// ---------------------------------------------------------------------------
__global__ __launch_bounds__(256) void gat_attn_kernel(
    const float* __restrict__ outP, const float* __restrict__ sself,
    const int* __restrict__ nbr, const float* __restrict__ a,
    float* __restrict__ out) {
  const int lane = threadIdx.x & 31;
  const int node = blockIdx.x * (blockDim.x >> 5) + (threadIdx.x >> 5);
  if (node >= N_NODES) return;

  int nb = nbr[node * KNN + lane];
  v4f o0 = *(const v4f*)(outP + nb * 8);
  v4f o1 = *(const v4f*)(outP + nb * 8 + 4);  // o1.w = neighbor pollutant

  float s = sself[node] + o0.x * a[7] + o0.y * a[8] + o0.z * a[9] +
            o0.w * a[10] + o1.x * a[11] + o1.y * a[12] + o1.z * a[13];

  float mx = s;
#pragma unroll
  for (int off = 16; off > 0; off >>= 1)
    mx = fmaxf(mx, __shfl_xor(mx, off, 32));
  float e = expf(s - mx);
  float sum = e;
#pragma unroll
  for (int off = 16; off > 0; off >>= 1) sum += __shfl_xor(sum, off, 32);
  float att = e / sum;

  float g[8] = {att * o0.x, att * o0.y, att * o0.z, att * o0.w,
                att * o1.x, att * o1.y, att * o1.z, att * o1.w};
#pragma unroll
  for (int off = 16; off > 0; off >>= 1) {
#pragma unroll
    for (int d = 0; d < 8; ++d) g[d] += __shfl_xor(g[d], off, 32);
  }

  if (lane == 0) {
#pragma unroll
    for (int d = 0; d < 7; ++d) {
      float v = outP[node * 8 + d];
      out[node * 15 + d] = v;
      out[N_NODES * 15 + node * 15 + d] = v;
    }
#pragma unroll
    for (int d = 0; d < 8; ++d) {
      out[node * 15 + 7 + d] = g[d];
      out[N_NODES * 15 + node * 15 + 7 + d] = g[d];
    }
  }
  if (node == N_NODES - 1) out[2 * N_NODES * 15 + lane] = att;
}

// ---------------------------------------------------------------------------
extern "C" void kernel_launch(void* const* d_in, const int* in_sizes, int n_in,
                              void* d_out, int out_size, void* d_ws,
                              size_t ws_size, hipStream_t stream) {
  const float* x = (const float*)d_in[0];  // [8192, 8]
  const float* W = (const float*)d_in[1];  // [7, 7]
  const float* a = (const float*)d_in[2];  // [14]

  // Workspace layout.
  float* ws = (float*)d_ws;
  float* featP = ws;                   // 8192*8 f32
  float* outP = featP + N_NODES * 8;   // 8192*8 f32
  float* sq = outP + N_NODES * 8;      // 8192 f32
  float* sself = sq + N_NODES;         // 8192 f32
  int* nbr = (int*)(sself + N_NODES);  // 8192*32 i32
  float* pval = (float*)(nbr + N_NODES * KNN);

  // Column-split factor: 4 quadruples wave count (256 -> 1024 waves) for
  // latency hiding; fall back to 1 if scratch is tight (deterministic).
  const size_t fixedBytes = ((size_t)N_NODES * (8 + 8 + 1 + 1) + (size_t)N_NODES * KNN) * 4;
  int nchunks = 4;
  if (ws_size < fixedBytes + (size_t)N_NODES * nchunks * KNN * 8) nchunks = 1;
  int* pidx = (int*)(pval + (size_t)N_NODES * nchunks * KNN);
  const int colsPerChunk = N_NODES / nchunks;

  float* out = (float*)d_out;

  gat_prep_kernel<<<N_NODES / 256, 256, 0, stream>>>(x, W, a, featP, outP, sq,
                                                     sself);
  gat_knn_kernel<<<(N_NODES / 128) * nchunks, 128, 0, stream>>>(
      featP, sq, pval, pidx, nchunks, colsPerChunk);
  gat_merge_kernel<<<N_NODES / 256, 256, 0, stream>>>(pval, pidx, nbr,
                                                      nchunks);
  gat_attn_kernel<<<N_NODES / 8, 256, 0, stream>>>(outP, sself, nbr, a, out);
}